// CascadedAttention_60825326846600
// MI455X (gfx1250) — compile-verified
//
#include <hip/hip_runtime.h>
#include <math.h>

#define BB 128
#define TT 75
#define DD 512
#define VV 28
#define GG (3 * VV)   // 84

typedef __attribute__((ext_vector_type(2))) float v2f;
typedef __attribute__((ext_vector_type(8))) float v8f;

// ---------------------------------------------------------------------------
// UaH[M,N] = X[M,K] @ Ua[K,N] + Ba2[N],  M = B*T = 9600, K = N = D = 512.
// f32 WMMA: V_WMMA_F32_16X16X4_F32, one 16x16 output tile per wave32,
// K stepped by 4 (128 WMMA per tile). 8 waves (8 tiles) per 256-thread block.
// (Verified in round 1: lowers to v_wmma_f32_16x16x4_f32 with loadcnt
//  software pipelining.)
// ---------------------------------------------------------------------------
__global__ __launch_bounds__(256) void uah_gemm_wmma(
        const float* __restrict__ X,    // [9600, 512]
        const float* __restrict__ Ua,   // [512, 512]
        const float* __restrict__ Ba2,  // [512]
        float* __restrict__ UaH) {      // [9600, 512]
    const int lane = threadIdx.x & 31;
    const int wave = threadIdx.x >> 5;
    const int tile = blockIdx.x * 8 + wave;          // 0 .. 19199
    const int ntilesN = DD / 16;                     // 32
    const int tileM = tile / ntilesN;                // 0 .. 599
    const int tileN = tile - tileM * ntilesN;

    const int half = lane >> 4;                      // 0 or 1
    const int m    = lane & 15;

    const float* xrow = X + (size_t)(tileM * 16 + m) * DD;
    const int    colB = tileN * 16 + m;

    v8f c = {};
    for (int k0 = 0; k0 < DD; k0 += 4) {
        const int ka = k0 + 2 * half;
        v2f a, b;
        a.x = xrow[ka];
        a.y = xrow[ka + 1];
        b.x = Ua[(size_t)ka * DD + colB];
        b.y = Ua[(size_t)(ka + 1) * DD + colB];
        c = __builtin_amdgcn_wmma_f32_16x16x4_f32(
                false, a, false, b, (short)0, c, false, false);
    }

    const int   col  = tileN * 16 + m;
    const float bias = Ba2[col];
#pragma unroll
    for (int r = 0; r < 8; ++r) {
        const int row = tileM * 16 + r + half * 8;
        UaH[(size_t)row * DD + col] = c[r] + bias;
    }
}

// ---------------------------------------------------------------------------
// Fused 75-step scan. One block per batch row b; the carry (state, pred) and
// per-step context live entirely in LDS. All step-invariant weights are
// staged once into CDNA5's 320 KB LDS (~284 KB dynamic + ~25 KB static), so
// the only per-step global traffic is streaming UaH[b] and x[b] (L2-resident).
// 512 threads = 16 wave32.
// ---------------------------------------------------------------------------
// dynamic LDS layout (floats):
//   sgk  [D*G]  = 43008
//   sWa  [V*D]  = 14336
//   sCo  [D*V]  = 14336
//   sVa  [D]    =   512
//   sBa1 [D]    =   512
// total 72704 floats = 290816 bytes
#define SMEM_FLOATS (DD * GG + VV * DD + DD * VV + DD + DD)

__global__ __launch_bounds__(512) void scan_kernel(
        const float* __restrict__ x,     // [B,T,D]
        const float* __restrict__ UaH,   // [B,T,D]
        const float* __restrict__ Wa,    // [V,D]
        const float* __restrict__ Va,    // [D]
        const float* __restrict__ Ba1,   // [D]
        const float* __restrict__ Ba3,   // [1]
        const float* __restrict__ gk,    // [D,3V]
        const float* __restrict__ grk,   // [V,3V]
        const float* __restrict__ gbias, // [2,3V]
        const float* __restrict__ Wo,    // [V]
        const float* __restrict__ Uo,    // [V,V]
        const float* __restrict__ Co,    // [D,V]
        const float* __restrict__ Bo,    // [V]
        const float* __restrict__ emb,   // [V,V]
        float* __restrict__ out) {       // [B,T,V]
    extern __shared__ float smem[];
    float* sgk  = smem;                        // [D*G]
    float* sWa  = sgk  + DD * GG;              // [V*D]
    float* sCo  = sWa  + VV * DD;              // [D*V]
    float* sVa  = sCo  + DD * VV;              // [D]
    float* sBa1 = sVa  + DD;                   // [D]

    __shared__ float sWaS[DD];
    __shared__ float sctx[DD];
    __shared__ float sw[TT];
    __shared__ float sstate[VV], spred[VV], snew[VV], slog[VV];
    __shared__ float sxmp[4 * GG];             // xm partials (4 K-chunks)
    __shared__ float sxm[GG], shm[GG];
    __shared__ float sCoCp[16 * VV];           // CoC partials (16 K-chunks)
    __shared__ float sgb[2 * GG];
    __shared__ float sgrk[VV * GG];
    __shared__ float sUo[VV * VV], semb[VV * VV];
    __shared__ float sWo[VV], sBo[VV];
    __shared__ float smax, ssum, sba3;

    const int b    = blockIdx.x;
    const int tid  = threadIdx.x;
    const int lane = tid & 31;
    const int wave = tid >> 5;                 // 0..15

    // ---- stage all step-invariant weights into LDS (once per block) ----
    for (int i = tid; i < DD * GG; i += 512) sgk[i] = gk[i];
    for (int i = tid; i < VV * DD; i += 512) sWa[i] = Wa[i];
    for (int i = tid; i < DD * VV; i += 512) sCo[i] = Co[i];
    for (int i = tid; i < DD; i += 512) { sVa[i] = Va[i]; sBa1[i] = Ba1[i]; }
    for (int i = tid; i < VV * GG; i += 512) sgrk[i] = grk[i];
    for (int i = tid; i < VV * VV; i += 512) { sUo[i] = Uo[i]; semb[i] = emb[i]; }
    for (int i = tid; i < 2 * GG; i += 512) sgb[i] = gbias[i];
    if (tid < VV) {
        sWo[tid] = Wo[tid];
        sBo[tid] = Bo[tid];
        sstate[tid] = 0.0f;    // carry init
        spred[tid]  = 0.0f;
    }
    if (tid == 0) sba3 = Ba3[0];
    __syncthreads();

    const float* xb   = x   + (size_t)b * TT * DD;
    const float* uahb = UaH + (size_t)b * TT * DD;

    for (int t = 0; t < TT; ++t) {
        // ---- WaS = prev_state @ Wa + Ba1 (one d per thread) ----
        {
            const int d = tid;
            float acc = sBa1[d];
#pragma unroll
            for (int v = 0; v < VV; ++v) acc += sstate[v] * sWa[v * DD + d];
            sWaS[d] = acc;
        }
        __syncthreads();

        // ---- scores[t2] = sum_d tanh(UaH+WaS)*Va + Ba3 (waves stripe t2) --
        for (int t2 = wave; t2 < TT; t2 += 16) {
            const float* u = uahb + (size_t)t2 * DD;
            if (t2 + 16 < TT)                            // next row this wave
                __builtin_prefetch(u + 16 * DD, 0, 1);   // global_prefetch_b8
            float acc = 0.0f;
            for (int d = lane; d < DD; d += 32)
                acc += tanhf(u[d] + sWaS[d]) * sVa[d];
#pragma unroll
            for (int off = 16; off > 0; off >>= 1)
                acc += __shfl_down(acc, off, 32);
            if (lane == 0) sw[t2] = acc + sba3;
        }
        __syncthreads();

        // ---- softmax over T (wave 0, wave32-wide reductions) ----
        if (wave == 0) {
            float mx = -1e30f;
            for (int t2 = lane; t2 < TT; t2 += 32) mx = fmaxf(mx, sw[t2]);
#pragma unroll
            for (int off = 16; off > 0; off >>= 1)
                mx = fmaxf(mx, __shfl_down(mx, off, 32));
            mx = __shfl(mx, 0, 32);
            float s = 0.0f;
            for (int t2 = lane; t2 < TT; t2 += 32) {
                float e = __expf(sw[t2] - mx);
                sw[t2] = e;
                s += e;
            }
#pragma unroll
            for (int off = 16; off > 0; off >>= 1)
                s += __shfl_down(s, off, 32);
            s = __shfl(s, 0, 32);
            const float inv = 1.0f / s;
            for (int t2 = lane; t2 < TT; t2 += 32) sw[t2] *= inv;
        }
        __syncthreads();

        // ---- ctx[d] = sum_t x[b,t,d]*w[t] (one d per thread) ----
        {
            const int d = tid;
            float acc = 0.0f;
            const float* xd = xb + d;
            for (int t2 = 0; t2 < TT; ++t2) acc += xd[(size_t)t2 * DD] * sw[t2];
            sctx[d] = acc;
        }
        __syncthreads();

        // ---- xm partials (4 chunks x 84 cols), hm, CoC partials ----
        {
            const int chunk = tid >> 7;        // 0..3
            const int j     = tid & 127;
            if (j < GG) {
                float p = 0.0f;
                const int d0 = chunk * 128;
                for (int d = d0; d < d0 + 128; ++d) p += sctx[d] * sgk[d * GG + j];
                sxmp[chunk * GG + j] = p;
            }
            if (tid < GG) {
                float hm = sgb[GG + tid];
#pragma unroll
                for (int v = 0; v < VV; ++v) hm += sstate[v] * sgrk[v * GG + tid];
                shm[tid] = hm;
            }
            const int c2 = tid >> 5;           // 0..15
            const int j2 = tid & 31;
            if (j2 < VV) {
                float p = 0.0f;
                const int d0 = c2 * 32;
                for (int d = d0; d < d0 + 32; ++d) p += sctx[d] * sCo[d * VV + j2];
                sCoCp[c2 * VV + j2] = p;
            }
        }
        __syncthreads();

        if (tid < GG)
            sxm[tid] = sgb[tid] + sxmp[tid] + sxmp[GG + tid]
                     + sxmp[2 * GG + tid] + sxmp[3 * GG + tid];
        __syncthreads();

        // ---- gates + logits (tid < V) ----
        if (tid < VV) {
            const float z  = 1.0f / (1.0f + __expf(-(sxm[tid] + shm[tid])));
            const float r  = 1.0f / (1.0f + __expf(-(sxm[VV + tid] + shm[VV + tid])));
            const float hh = tanhf(sxm[2 * VV + tid] + r * shm[2 * VV + tid]);
            snew[tid] = z * sstate[tid] + (1.0f - z) * hh;

            int idx = (int)spred[tid];
            idx = idx < 0 ? 0 : (idx > VV - 1 ? VV - 1 : idx);
            float woy = 0.0f;
#pragma unroll
            for (int k = 0; k < VV; ++k) woy += semb[idx * VV + k] * sWo[k];
            float uoh = 0.0f;
#pragma unroll
            for (int v = 0; v < VV; ++v) uoh += sstate[v] * sUo[v * VV + tid];
            float coc = 0.0f;
#pragma unroll
            for (int c = 0; c < 16; ++c) coc += sCoCp[c * VV + tid];
            slog[tid] = woy + uoh + coc + sBo[tid];
        }
        __syncthreads();

        // ---- softmax over V=28 + write output + update carry ----
        if (tid == 0) {
            float mx = slog[0];
            for (int j = 1; j < VV; ++j) mx = fmaxf(mx, slog[j]);
            float s = 0.0f;
            for (int j = 0; j < VV; ++j) s += __expf(slog[j] - mx);
            smax = mx;
            ssum = s;
        }
        __syncthreads();
        if (tid < VV) {
            const float p = __expf(slog[tid] - smax) / ssum;
            out[((size_t)b * TT + t) * VV + tid] = p;
            spred[tid]  = p;
            sstate[tid] = snew[tid];
        }
        __syncthreads();   // carry visible before next step's WaS phase
    }
}

// ---------------------------------------------------------------------------
extern "C" void kernel_launch(void* const* d_in, const int* in_sizes, int n_in,
                              void* d_out, int out_size, void* d_ws, size_t ws_size,
                              hipStream_t stream) {
    (void)in_sizes; (void)n_in; (void)out_size; (void)ws_size;

    const float* x     = (const float*)d_in[0];   // [B,T,D]
    const float* Wa    = (const float*)d_in[1];   // [V,D]
    const float* Ua    = (const float*)d_in[2];   // [D,D]
    const float* Va    = (const float*)d_in[3];   // [D,1]
    const float* Ba1   = (const float*)d_in[4];   // [1,D]
    const float* Ba2   = (const float*)d_in[5];   // [1,D]
    const float* Ba3   = (const float*)d_in[6];   // [1,1]
    const float* gk    = (const float*)d_in[7];   // [D,3V]
    const float* grk   = (const float*)d_in[8];   // [V,3V]
    const float* gbias = (const float*)d_in[9];   // [2,3V]
    const float* Wo    = (const float*)d_in[10];  // [V,1]
    const float* Uo    = (const float*)d_in[11];  // [V,V]
    const float* Co    = (const float*)d_in[12];  // [D,V]
    const float* Bo    = (const float*)d_in[13];  // [1,V]
    const float* emb   = (const float*)d_in[14];  // [V,V]
    float* out = (float*)d_out;

    float* UaH = (float*)d_ws;                    // B*T*D floats

    // UaH = x @ Ua + Ba2 (f32 WMMA): 600*32 tiles, 8 tiles/block
    {
        const int tiles = (BB * TT / 16) * (DD / 16);  // 19200
        uah_gemm_wmma<<<tiles / 8, 256, 0, stream>>>(x, Ua, Ba2, UaH);
    }

    // Whole 75-step scan fused into one kernel: one block per batch row,
    // carry in LDS, all weights staged in CDNA5's 320 KB LDS.
    scan_kernel<<<BB, 512, SMEM_FLOATS * sizeof(float), stream>>>(
        x, UaH, Wa, Va, Ba1, Ba3, gk, grk, gbias, Wo, Uo, Co, Bo, emb, out);
}